// LateInteraction_2954937500508
// MI455X (gfx1250) — compile-verified
//
#include <hip/hip_runtime.h>
#include <math.h>

typedef float v8f __attribute__((ext_vector_type(8)));
typedef float v2f __attribute__((ext_vector_type(2)));

// Reference dims (fixed by setup_inputs)
constexpr int Bb = 32, Lq = 32, Nn = 100, Ld = 512, Dd = 128;

// ---------------- Kernel 1: L2-normalize x1 rows, apply x1_mask -------------
__global__ __launch_bounds__(128) void li_normalize_x1(const float* __restrict__ x1,
                                                       const float* __restrict__ x1_mask,
                                                       float* __restrict__ x1n) {
    const int row = blockIdx.x;          // 0 .. B*Lq-1
    const int tid = threadIdx.x;         // 0 .. 127 (== D)
    float v = x1[(size_t)row * Dd + tid];
    float ss = v * v;
    #pragma unroll
    for (int off = 16; off > 0; off >>= 1) ss += __shfl_xor(ss, off, 32);
    __shared__ float s[4];
    const int wid = tid >> 5, lane = tid & 31;
    if (lane == 0) s[wid] = ss;
    __syncthreads();
    const float total = s[0] + s[1] + s[2] + s[3];
    const float inv = 1.0f / fmaxf(sqrtf(total), 1e-12f);
    x1n[(size_t)row * Dd + tid] = v * inv * x1_mask[row];
}

// ---------------- Kernel 2: MaxSim per (b, n) via V_WMMA_F32_16X16X4_F32 ----
// Block = 256 threads (8 waves). Wave w: q-tile = w&1, l-tiles = (w>>1) + 4*j.
// Raw dot S = x1n * x2^T accumulated over K=128 in 32 WMMA(16x16x4) steps.
// The whole 16x128 B tile (8 KB) is preloaded into 32 distinct v2f register
// pairs so all 32 global_load_b64 are in flight simultaneously (no
// load->wait->wmma serialization). Per-column 1/||x2_l|| is computed inline
// from the same fragments and applied after accumulation (equivalent to
// normalizing x2 first, since max over l commutes with positive column scaling).
__global__ __launch_bounds__(256) void li_maxsim(const float* __restrict__ x1n,
                                                 const float* __restrict__ x2,
                                                 const float* __restrict__ x2_mask,
                                                 float* __restrict__ out) {
    const int bn     = blockIdx.x;        // 0 .. B*N-1
    const int b      = bn / Nn;
    const int wave   = threadIdx.x >> 5;
    const int lane   = threadIdx.x & 31;
    const int laneLo = lane & 15;
    const int hi     = lane >> 4;         // 0: K pair {4k,4k+1}, 1: {4k+2,4k+3}
    const int qt     = wave & 1;

    // Preload A fragments: 16x128 slice of normalized queries -> 32 x v2f regs.
    // A layout (16x4 f32): lanes 0-15 hold M=0..15 with K=0,1; lanes 16-31 K=2,3.
    const float* qrow = x1n + ((size_t)b * Lq + qt * 16 + laneLo) * Dd + hi * 2;
    v2f a[32];
    #pragma unroll
    for (int kk = 0; kk < 32; ++kk) a[kk] = *(const v2f*)(qrow + 4 * kk);

    const float* x2bn = x2 + (size_t)bn * Ld * Dd;
    const float* mrow = x2_mask + (size_t)bn * Ld;

    float vmax[8];
    #pragma unroll
    for (int r = 0; r < 8; ++r) vmax[r] = -INFINITY;

    for (int lt = (wave >> 1); lt < Ld / 16; lt += 4) {
        const int l0 = lt * 16;
        // B layout (4x16 f32): lane c<16 holds (K=4k+{0,1}, N=c); lane c+16 (K=4k+{2,3}, N=c)
        const float* brow = x2bn + (size_t)(l0 + laneLo) * Dd + hi * 2;

        // Stage the full B tile in registers: 32 independent b64 loads in flight.
        v2f bf[32];
        #pragma unroll
        for (int kk = 0; kk < 32; ++kk) bf[kk] = *(const v2f*)(brow + 4 * kk);

        v8f c = {};
        float ssq = 0.0f;
        #pragma unroll
        for (int kk = 0; kk < 32; ++kk) {
            ssq += bf[kk][0] * bf[kk][0] + bf[kk][1] * bf[kk][1];
            c = __builtin_amdgcn_wmma_f32_16x16x4_f32(
                    /*neg_a=*/false, a[kk], /*neg_b=*/false, bf[kk],
                    /*c_mod=*/(short)0, c, /*reuse_a=*/false, /*reuse_b=*/false);
        }
        // Column N=c sum-of-squares split between lane c and lane c+16 -> combine.
        ssq += __shfl_xor(ssq, 16, 32);
        const float scale = (1.0f / fmaxf(sqrtf(ssq), 1e-12f)) * mrow[l0 + laneLo];
        // C layout: lane c<16 -> (M=r, N=c); lane c+16 -> (M=8+r, N=c). Column scale
        // is per-lane, so running max over l is a per-lane elementwise max.
        #pragma unroll
        for (int r = 0; r < 8; ++r) vmax[r] = fmaxf(vmax[r], c[r] * scale);
    }

    // Max over the 16 columns held by each half-wave (rows stay per-lane-slot).
    #pragma unroll
    for (int off = 1; off < 16; off <<= 1) {
        #pragma unroll
        for (int r = 0; r < 8; ++r) vmax[r] = fmaxf(vmax[r], __shfl_xor(vmax[r], off, 32));
    }

    // red[wave][0..7] = rows qt*16+0..7 max, red[wave][8..15] = rows qt*16+8..15 max
    __shared__ float red[8][16];
    if (lane == 0) {
        #pragma unroll
        for (int r = 0; r < 8; ++r) red[wave][r] = vmax[r];
    } else if (lane == 16) {
        #pragma unroll
        for (int r = 0; r < 8; ++r) red[wave][8 + r] = vmax[r];
    }
    __syncthreads();

    if (wave == 0) {
        const int q  = lane;        // query row 0..31
        const int qq = q >> 4;      // which q-tile
        const int r  = q & 15;
        float m = -INFINITY;
        #pragma unroll
        for (int j = 0; j < 4; ++j) m = fmaxf(m, red[2 * j + qq][r]);  // waves w&1==qq
        #pragma unroll
        for (int off = 16; off > 0; off >>= 1) m += __shfl_xor(m, off, 32);
        if (lane == 0) out[bn] = m;
    }
}

extern "C" void kernel_launch(void* const* d_in, const int* in_sizes, int n_in,
                              void* d_out, int out_size, void* d_ws, size_t ws_size,
                              hipStream_t stream) {
    (void)in_sizes; (void)n_in; (void)out_size; (void)ws_size;
    const float* x1      = (const float*)d_in[0];
    const float* x2      = (const float*)d_in[1];
    const float* x1_mask = (const float*)d_in[2];
    const float* x2_mask = (const float*)d_in[3];
    float* x1n = (float*)d_ws;   // B*Lq*D floats = 512 KB scratch

    li_normalize_x1<<<Bb * Lq, 128, 0, stream>>>(x1, x1_mask, x1n);
    li_maxsim<<<Bb * Nn, 256, 0, stream>>>(x1n, x2, x2_mask, (float*)d_out);
}